// KNNGroup_1468878815326
// MI455X (gfx1250) — compile-verified
//
#include <hip/hip_runtime.h>

// KNN (K=32) grouping: d2 via V_WMMA_F32_16X16X4_F32, wave-parallel top-k, coalesced gather-write.
//
// Shapes: query [8,8192,3], support [8,2048,3], features [8,64,2048]
// Outputs (concat): grouped_xyz [8,3,8192,32], grouped_features [8,64,8192,32]

#define NP   2048
#define MQ   8192
#define KNN  32
#define CH   64
#define D2_STRIDE 2080   // 2048 + 32 dwords

typedef __attribute__((ext_vector_type(2))) float v2f;
typedef __attribute__((ext_vector_type(8))) float v8f;

// monotonic unsigned encoding of float (handles negatives from rounding)
__device__ __forceinline__ unsigned enc_f32(float v) {
    unsigned b = __float_as_uint(v);
    return b ^ ((b & 0x80000000u) ? 0xFFFFFFFFu : 0x80000000u);
}

__global__ __launch_bounds__(128) void knn_group_kernel(
    const float* __restrict__ query,     // [B,MQ,3]
    const float* __restrict__ support,   // [B,NP,3]
    const float* __restrict__ features,  // [B,CH,NP]
    float* __restrict__ out_xyz,         // [B,3,MQ,KNN]
    float* __restrict__ out_feat)        // [B,CH,MQ,KNN]
{
    __shared__ float s_d2[16 * D2_STRIDE];            // 133,120 B
    __shared__ float s_sx[NP], s_sy[NP], s_sz[NP];    //  24,576 B
    __shared__ float s_qx[16], s_qy[16], s_qz[16], s_q2[16];
    __shared__ int   s_idx[16 * KNN];

    const int t    = threadIdx.x;
    const int lane = t & 31;
    const int wv   = t >> 5;          // 4 waves per block
    const int b    = blockIdx.y;      // batch
    const int m0   = blockIdx.x * 16; // query tile base

    // ---- stage 0: load support SoA + query tile into LDS ----
    const float* sp = support + (size_t)b * NP * 3;
    for (int n = t; n < NP; n += 128) {
        s_sx[n] = sp[n * 3 + 0];
        s_sy[n] = sp[n * 3 + 1];
        s_sz[n] = sp[n * 3 + 2];
    }
    if (t < 16) {
        const float* q = query + ((size_t)b * MQ + m0 + t) * 3;
        float x = q[0], y = q[1], z = q[2];
        s_qx[t] = x; s_qy[t] = y; s_qz[t] = z;
        s_q2[t] = x * x + y * y + z * z;
    }
    __syncthreads();

    // ---- stage 1: d2[16][2048] via WMMA f32 16x16x4 ----
    // A[m,:] = (-2qx, -2qy, -2qz, 1)   (16x4)
    // B[:,n] = ( sx ,  sy ,  sz , s2)  (4x16)
    // C[m,n] = q2[m]
    // D = A*B + C = q2 + s2 - 2 q.s = ||q-s||^2
    const int hi = lane >> 4;   // A: K-pair select; B: K-pair select; C/D: M-half select
    const int ml = lane & 15;   // A: row M; B: col N

    v2f a;
    a.x = -2.0f * (hi ? s_qz[ml] : s_qx[ml]);          // K=2 : K=0
    a.y = hi ? 1.0f : (-2.0f * s_qy[ml]);              // K=3 : K=1
    v8f c;
#pragma unroll
    for (int r = 0; r < 8; ++r) c[r] = s_q2[r + 8 * hi];

    // each wave covers NP/4 = 512 support cols -> 32 WMMA tiles
    for (int it = 0; it < NP / (16 * 4); ++it) {
        const int n0 = wv * (NP / 4) + it * 16;
        const int n  = n0 + ml;
        float sx = s_sx[n], sy = s_sy[n], sz = s_sz[n];
        v2f bb;
        bb.x = hi ? sz : sx;                               // K=2 : K=0
        bb.y = hi ? (sx * sx + sy * sy + sz * sz) : sy;    // K=3 : K=1
        v8f d = __builtin_amdgcn_wmma_f32_16x16x4_f32(
            /*neg_a=*/false, a, /*neg_b=*/false, bb,
            /*c_mod=*/(short)0, c, /*reuse_a=*/false, /*reuse_b=*/false);
#pragma unroll
        for (int r = 0; r < 8; ++r)
            s_d2[(r + 8 * hi) * D2_STRIDE + n0 + ml] = d[r];
    }
    __syncthreads();

    // ---- stage 2: top-32 per query row (wave-parallel argmin extraction) ----
    // Lane L owns row entries {L, L+32, ...} (disjoint slices -> no cross-lane LDS deps).
    const float INF = __uint_as_float(0x7f800000u);
    for (int qq = 0; qq < 4; ++qq) {
        const int m = wv * 4 + qq;
        float* row = &s_d2[m * D2_STRIDE];

        unsigned long long bk = ~0ull;
        for (int j = 0; j < NP / 32; ++j) {
            int ii = lane + 32 * j;
            unsigned long long kk =
                ((unsigned long long)enc_f32(row[ii]) << 32) | (unsigned)ii;
            bk = kk < bk ? kk : bk;
        }
        for (int k = 0; k < KNN; ++k) {
            unsigned long long red = bk;
#pragma unroll
            for (int off = 16; off > 0; off >>= 1) {
                unsigned long long o = __shfl_xor(red, off, 32);
                red = o < red ? o : red;
            }
            if (bk == red) {                 // unique winner (idx in low bits)
                int widx = (int)(red & 0xFFFFFFFFull);
                s_idx[m * KNN + k] = widx;
                row[widx] = INF;             // invalidate own slice entry
                bk = ~0ull;                  // rescan own slice
                for (int j = 0; j < NP / 32; ++j) {
                    int ii = lane + 32 * j;
                    unsigned long long kk =
                        ((unsigned long long)enc_f32(row[ii]) << 32) | (unsigned)ii;
                    bk = kk < bk ? kk : bk;
                }
            }
        }
    }
    __syncthreads();

    // ---- stage 3: write grouped_xyz (relative) ----
    for (int p = t; p < 3 * 16 * KNN; p += 128) {
        int cc = p / (16 * KNN);
        int rr = p % (16 * KNN);
        int m  = rr / KNN, k = rr % KNN;
        int id = s_idx[m * KNN + k];
        float sv = (cc == 0) ? s_sx[id] : (cc == 1) ? s_sy[id] : s_sz[id];
        float qv = (cc == 0) ? s_qx[m] : (cc == 1) ? s_qy[m] : s_qz[m];
        out_xyz[(((size_t)b * 3 + cc) * MQ + m0 + m) * KNN + k] = sv - qv;
    }

    // ---- stage 4: write grouped_features (gather from L2-resident features) ----
    const float* fb = features + (size_t)b * CH * NP;
    for (int cc = 0; cc < CH; ++cc) {
        const float* frow  = fb + (size_t)cc * NP;
        const size_t obase = (((size_t)b * CH + cc) * MQ + m0) * KNN;
        for (int p = t; p < 16 * KNN; p += 128) {       // p = m*32 + k, contiguous stores
            out_feat[obase + p] = frow[s_idx[p]];
        }
    }
}

extern "C" void kernel_launch(void* const* d_in, const int* in_sizes, int n_in,
                              void* d_out, int out_size, void* d_ws, size_t ws_size,
                              hipStream_t stream) {
    (void)in_sizes; (void)n_in; (void)out_size; (void)d_ws; (void)ws_size;
    const float* query    = (const float*)d_in[0];
    const float* support  = (const float*)d_in[1];
    const float* features = (const float*)d_in[2];
    float* out_xyz  = (float*)d_out;
    float* out_feat = out_xyz + (size_t)8 * 3 * MQ * KNN;   // 6,291,456 floats

    dim3 grid(MQ / 16, 8);   // 512 query tiles x 8 batches
    knn_group_kernel<<<grid, 128, 0, stream>>>(query, support, features,
                                               out_xyz, out_feat);
}